// MaskedTemporalVisionTransformer_32667521253559
// MI455X (gfx1250) — compile-verified
//
#include <hip/hip_runtime.h>
#include <stdint.h>

// out[b, 0, h, w] = min over c of x[b, T-1, c, h, w]
// x: [B=4, T=8, C=32, H=512, W=512] f32  -> out: [4, 1, 512, 512] f32
//
// HBM-bound streaming reduction (128 MiB read, 4 MiB write ~ 5.9 us at
// 23.3 TB/s). No matrix math -> WMMA inapplicable; the CDNA5-specific path
// is the async global->LDS data mover (global_load_async_to_lds_b128 +
// s_wait_asynccnt), here used as a depth-4 software pipeline. A slot is only
// re-used for channel c+4 after s_wait_dscnt 0 confirms the ds_load readback
// of channel c has retired (async LDS writes are NOT ordered vs DS reads).

namespace {

constexpr int B = 4, T = 8, C = 32, H = 512, W = 512;
constexpr int HWPIX = H * W;                   // 262144 pixels per channel plane
constexpr int THREADS = 256;                   // 8 wave32 per block
constexpr int VEC = 4;                         // float4 per thread
constexpr int TILE = THREADS * VEC;            // 1024 pixels per block
constexpr int TILES_PER_BATCH = HWPIX / TILE;  // 256
constexpr int DEPTH = 4;                       // async pipeline depth (ring)

#define ASYNC_COPY_B128(ldsaddr, gaddr)                                   \
    asm volatile("global_load_async_to_lds_b128 %0, %1, off"              \
                 :: "v"(ldsaddr), "v"(gaddr) : "memory")

__global__ __launch_bounds__(THREADS) void min_over_c_kernel(
    const float* __restrict__ x, float* __restrict__ out)
{
    // Depth-4 ring of 4 KiB staging buffers (16 KiB of the 320 KiB WGP LDS).
    __shared__ __align__(16) float smem[DEPTH][THREADS * VEC];

    const int tid  = threadIdx.x;
    const int b    = blockIdx.x / TILES_PER_BATCH;
    const int tile = blockIdx.x % TILES_PER_BATCH;
    const long long pix = (long long)tile * TILE + (long long)tid * VEC;

    // Base of (b, t = T-1, c = 0) plane, plus this thread's pixel offset.
    const float* base =
        x + ((long long)(b * T + (T - 1)) * C) * (long long)HWPIX + pix;

    // Per-lane LDS byte offsets (low 32 bits of the flat shared address are
    // the wave-relative LDS offset on AMDGPU).
    unsigned lds[DEPTH];
    #pragma unroll
    for (int i = 0; i < DEPTH; ++i)
        lds[i] = (unsigned)(uintptr_t)&smem[i][tid * VEC];

    // Prologue: stage channels 0..3 into slots 0..3 (16 B per lane each).
    #pragma unroll
    for (int c = 0; c < DEPTH; ++c) {
        unsigned long long ga = (unsigned long long)(uintptr_t)
            (base + (long long)c * HWPIX);
        ASYNC_COPY_B128(lds[c], ga);
    }

    float m0 = __builtin_inff(), m1 = m0, m2 = m0, m3 = m0;

    // Main loop: channels 0..C-DEPTH-1. Outstanding copies before the wait
    // are {c, c+1, c+2, c+3}; asynccnt<=3 => channel c landed (async copies
    // complete in order).
    #pragma unroll
    for (int c = 0; c < C - DEPTH; ++c) {
        asm volatile("s_wait_asynccnt 3" ::: "memory");

        // Each lane reads back exactly the 16 B slot its own async copy
        // wrote -> no workgroup barrier needed, only the per-wave counters.
        const float4 v =
            *reinterpret_cast<const float4*>(&smem[c & (DEPTH - 1)][tid * VEC]);
        m0 = fminf(m0, v.x);
        m1 = fminf(m1, v.y);
        m2 = fminf(m2, v.z);
        m3 = fminf(m3, v.w);

        // Close the WAR window: the ds_load above must retire before the
        // async copy below may overwrite the same ring slot.
        asm volatile("s_wait_dscnt 0" ::: "memory");

        unsigned long long ga = (unsigned long long)(uintptr_t)
            (base + (long long)(c + DEPTH) * HWPIX);
        ASYNC_COPY_B128(lds[c & (DEPTH - 1)], ga);
    }

    // Tail: drain the pipe; remaining outstanding copies shrink 3 -> 0.
    // No more slot reuse, so no dscnt waits are needed here.
    #define TAIL_STEP(cc, waitlit)                                            \
        do {                                                                  \
            asm volatile("s_wait_asynccnt " waitlit ::: "memory");            \
            const float4 v = *reinterpret_cast<const float4*>(               \
                &smem[(cc) & (DEPTH - 1)][tid * VEC]);                        \
            m0 = fminf(m0, v.x);                                              \
            m1 = fminf(m1, v.y);                                              \
            m2 = fminf(m2, v.z);                                              \
            m3 = fminf(m3, v.w);                                              \
        } while (0)

    TAIL_STEP(C - 4, "3");
    TAIL_STEP(C - 3, "2");
    TAIL_STEP(C - 2, "1");
    TAIL_STEP(C - 1, "0");
    #undef TAIL_STEP

    float4 r;
    r.x = m0; r.y = m1; r.z = m2; r.w = m3;
    *reinterpret_cast<float4*>(out + (long long)b * HWPIX + pix) = r;
}

} // namespace

extern "C" void kernel_launch(void* const* d_in, const int* in_sizes, int n_in,
                              void* d_out, int out_size, void* d_ws, size_t ws_size,
                              hipStream_t stream)
{
    (void)in_sizes; (void)n_in; (void)out_size; (void)d_ws; (void)ws_size;
    const float* x = (const float*)d_in[0];
    float* out = (float*)d_out;

    dim3 grid(B * TILES_PER_BATCH);   // 1024 blocks
    dim3 block(THREADS);              // 256 threads = 8 wave32
    min_over_c_kernel<<<grid, block, 0, stream>>>(x, out);
}